// TemporalAttention_24026047053981
// MI455X (gfx1250) — compile-verified
//
#include <hip/hip_runtime.h>
#include <math.h>
#include <stdint.h>

// Problem constants (from the reference)
#define S_LEN   2048
#define BATCH   128
#define DIM     64
#define NHEAD   8
#define HPAD    16          // heads padded to one WMMA M-tile
#define HIDW    32
#define CHUNK   128         // S rows per block (flash split)
#define NCHUNK  (S_LEN / CHUNK)
#define ESTR    68          // LDS row stride (floats) -> conflict-free transposed reads
#define ASTR    17          // att transpose staging stride

typedef float v2f __attribute__((ext_vector_type(2)));
typedef float v8f __attribute__((ext_vector_type(8)));

__device__ __forceinline__ v8f wmma_f32x4(v2f a, v2f b, v8f c) {
  // D = A(16x4 f32) * B(4x16 f32) + C(16x16 f32)
  return __builtin_amdgcn_wmma_f32_16x16x4_f32(false, a, false, b, (short)0, c,
                                               false, false);
}

// ---------------------------------------------------------------------------
// Kernel 1: per-batch prep.
//   q[h,b,:]   = hidden[b]*Wq_h + bq_h
//   qk[h,b,d]  = sum_e Wk[h,d,e]*q[h,b,e]      (score = enc . qk + sbias)
//   sbias[h,b] = sum_e q[h,b,e]*bk[h,e]
// Heads padded to 16 with zeros (pad heads yield finite, ignored softmax).
// ---------------------------------------------------------------------------
__global__ __launch_bounds__(64) void prep_kernel(
    const float* __restrict__ hidden, const float* __restrict__ Wq,
    const float* __restrict__ bq, const float* __restrict__ Wk,
    const float* __restrict__ bk, float* __restrict__ qk,
    float* __restrict__ sbias) {
  __shared__ float hS[DIM];
  __shared__ float qS[NHEAD][DIM];
  const int b = blockIdx.x;
  const int t = threadIdx.x;

  hS[t] = hidden[b * DIM + t];  // hidden is (1,B,D)
  __syncthreads();

  for (int idx = t; idx < NHEAD * DIM; idx += 64) {
    const int h = idx / DIM, e = idx % DIM;
    float s = bq[h * DIM + e];
    for (int d = 0; d < DIM; ++d) s += hS[d] * Wq[((size_t)h * DIM + d) * DIM + e];
    qS[h][e] = s;
  }
  __syncthreads();

  for (int idx = t; idx < HPAD * DIM; idx += 64) {
    const int h = idx / DIM, d = idx % DIM;
    float s = 0.0f;
    if (h < NHEAD)
      for (int e = 0; e < DIM; ++e) s += Wk[((size_t)h * DIM + d) * DIM + e] * qS[h][e];
    qk[((size_t)b * HPAD + h) * DIM + d] = s;
  }
  if (t < HPAD) {
    float s = 0.0f;
    if (t < NHEAD)
      for (int e = 0; e < DIM; ++e) s += qS[t][e] * bk[t * DIM + e];
    sbias[(size_t)b * HPAD + t] = s;
  }
}

// ---------------------------------------------------------------------------
// Kernel 2: flash attention partial over one (batch, S-chunk).  Single wave32.
//   scores^T tile (16 heads x 16 s) = qk (16xD) * enc^T (Dx16)   via WMMA f32
//   acc (16 heads x D)            += att (16x16) * enc (16xD)    via WMMA f32
//   l   (16 heads)                += att (16x16) * onescol       via WMMA f32
// enc chunk is staged with GLOBAL_LOAD_ASYNC_TO_LDS_B128 (ASYNCcnt path).
// ---------------------------------------------------------------------------
__global__ __launch_bounds__(32) void attn_partial_kernel(
    const float* __restrict__ enc,    // (S, B, D)
    const float* __restrict__ qk,     // (B, HPAD, D)
    const float* __restrict__ sbias,  // (B, HPAD)
    float* __restrict__ pm,           // (B, NCHUNK, HPAD)
    float* __restrict__ pl,           // (B, NCHUNK, HPAD)
    float* __restrict__ pacc) {       // (B, NCHUNK, HPAD, D)
  __shared__ __align__(16) float encS[CHUNK][ESTR];
  __shared__ float attT[HPAD][ASTR];

  const int lane = threadIdx.x;
  const int c = blockIdx.x;
  const int b = blockIdx.y;
  const int s0 = c * CHUNK;

  // Async-stage enc chunk (rows of batch b) into LDS: 128 x 64 f32 (padded).
  // LDS byte offset = low 32 bits of the generic (aperture) address.
  const unsigned encS_base = (unsigned)(uintptr_t)&encS[0][0];
  for (int i = lane; i < CHUNK * (DIM / 4); i += 32) {
    const int r = i >> 4;          // 16 float4 per row
    const int q4 = i & 15;
    const unsigned long long g =
        (unsigned long long)(const void*)(enc +
            (((size_t)(s0 + r)) * BATCH + b) * DIM + q4 * 4);
    const unsigned lds = encS_base + (unsigned)((r * ESTR + q4 * 4) * 4);
    asm volatile("global_load_async_to_lds_b128 %0, %1, off"
                 :: "v"(lds), "v"(g)
                 : "memory");
  }
  asm volatile("s_wait_asynccnt 0" ::: "memory");
  __syncthreads();

  // WMMA operand lane decode (f32 16x16x4):
  //   A: M = lane%16, VGPR0/1 hold K = kOff, kOff+1  (kOff = 0 or 2 per half)
  //   B: N = lane%16, VGPR0/1 hold K = kOff, kOff+1
  //   C: VGPR i holds M = i (lanes 0-15) / i+8 (lanes 16-31), N = lane%16
  const int hA = lane & 15;
  const int kOff = (lane & 16) ? 2 : 0;
  const int hBase = (lane & 16) ? 8 : 0;

  // Ones-column B operand: picks out row-sums (l) in C column 0.
  v2f onesB;
  onesB.x = (hA == 0) ? 1.0f : 0.0f;
  onesB.y = onesB.x;

  // Preload ONLY the A-operand pairs this lane actually feeds to WMMA:
  // per K-step of 4 it consumes qk[.., k+kOff] and qk[.., k+kOff+1].
  v2f qh[DIM / 4];  // 32 VGPRs instead of 64
  {
    const float* qrow = qk + ((size_t)b * HPAD + hA) * DIM + kOff;
#pragma unroll
    for (int j = 0; j < DIM / 4; ++j) {
      qh[j].x = qrow[j * 4];
      qh[j].y = qrow[j * 4 + 1];
    }
  }
  float sb[8];
#pragma unroll
  for (int i = 0; i < 8; ++i) sb[i] = sbias[(size_t)b * HPAD + hBase + i];

  float m_run[8];
  v8f acc[5];  // [0..3] = d-tiles of att*enc ; [4] = l (att row-sums, col 0)
#pragma unroll
  for (int i = 0; i < 8; ++i) m_run[i] = -3.0e38f;
#pragma unroll
  for (int n = 0; n < 5; ++n) acc[n] = (v8f){0.f, 0.f, 0.f, 0.f, 0.f, 0.f, 0.f, 0.f};

  for (int t = 0; t < CHUNK / 16; ++t) {
    const int sCol = t * 16 + (lane & 15);

    // scores^T tile: C(h, s) = sum_d qk[h,d] * enc[s,d]
    v8f sc = (v8f){0.f, 0.f, 0.f, 0.f, 0.f, 0.f, 0.f, 0.f};
#pragma unroll
    for (int k = 0; k < DIM; k += 4) {
      v2f bm; bm.x = encS[sCol][k + kOff];   bm.y = encS[sCol][k + kOff + 1];
      sc = wmma_f32x4(qh[k / 4], bm, sc);
    }

    // Online softmax max update per head (head = register index + lane half).
    float w[8], scl[8];
#pragma unroll
    for (int i = 0; i < 8; ++i) {
      const float v = sc[i] + sb[i];
      float mx = v;  // max over the 16 s-columns of this half
      mx = fmaxf(mx, __shfl_xor(mx, 1, 32));
      mx = fmaxf(mx, __shfl_xor(mx, 2, 32));
      mx = fmaxf(mx, __shfl_xor(mx, 4, 32));
      mx = fmaxf(mx, __shfl_xor(mx, 8, 32));
      const float m_new = fmaxf(m_run[i], mx);
      scl[i] = __expf(m_run[i] - m_new);
      m_run[i] = m_new;
      w[i] = __expf(v - m_new);
    }

    // Transpose att weights (C layout -> A layout) through LDS.
#pragma unroll
    for (int i = 0; i < 8; ++i) attT[hBase + i][lane & 15] = w[i];
    __syncthreads();

    // acc(h,d) = acc*scale + att(16x16)*enc_tile(16x64); acc[4] = l via ones.
#pragma unroll
    for (int n = 0; n < 5; ++n) {
      v8f a8 = acc[n];
#pragma unroll
      for (int i = 0; i < 8; ++i) a8[i] *= scl[i];
#pragma unroll
      for (int kk = 0; kk < 4; ++kk) {
        const int sK = t * 16 + kk * 4 + kOff;
        v2f a;  a.x = attT[hA][kk * 4 + kOff];
                a.y = attT[hA][kk * 4 + kOff + 1];
        v2f bm;
        if (n < 4) {
          bm.x = encS[sK][n * 16 + (lane & 15)];
          bm.y = encS[sK + 1][n * 16 + (lane & 15)];
        } else {
          bm = onesB;
        }
        a8 = wmma_f32x4(a, bm, a8);
      }
      acc[n] = a8;
    }
    __syncthreads();
  }

  // Emit partials.  l lives in C column 0 (lane%16 == 0) of acc[4].
  const size_t pbase = ((size_t)b * NCHUNK + c) * HPAD;
  if ((lane & 15) == 0) {
#pragma unroll
    for (int i = 0; i < 8; ++i) {
      pm[pbase + hBase + i] = m_run[i];
      pl[pbase + hBase + i] = acc[4][i];
    }
  }
#pragma unroll
  for (int n = 0; n < 4; ++n)
#pragma unroll
    for (int i = 0; i < 8; ++i)
      pacc[(pbase + hBase + i) * DIM + n * 16 + (lane & 15)] = acc[n][i];
}

// ---------------------------------------------------------------------------
// Kernel 3: merge chunk partials, project by Wv, run 512->32->1 MLP.
// ---------------------------------------------------------------------------
__global__ __launch_bounds__(256) void finalize_kernel(
    const float* __restrict__ pm, const float* __restrict__ pl,
    const float* __restrict__ pacc, const float* __restrict__ Wv,
    const float* __restrict__ bv, const float* __restrict__ W1,
    const float* __restrict__ b1, const float* __restrict__ W2,
    const float* __restrict__ b2, float* __restrict__ out) {
  __shared__ float Mh[NHEAD], Lh[NHEAD];
  __shared__ float ctxRaw[NHEAD][DIM];
  __shared__ float x[NHEAD * DIM];
  __shared__ float h1[HIDW];
  const int b = blockIdx.x;
  const int t = threadIdx.x;
  const size_t base = (size_t)b * NCHUNK;

  if (t < NHEAD) {
    float M = -3.0e38f;
    for (int c = 0; c < NCHUNK; ++c) M = fmaxf(M, pm[(base + c) * HPAD + t]);
    float L = 0.0f;
    for (int c = 0; c < NCHUNK; ++c)
      L += __expf(pm[(base + c) * HPAD + t] - M) * pl[(base + c) * HPAD + t];
    Mh[t] = M;
    Lh[t] = L;
  }
  __syncthreads();

  for (int idx = t; idx < NHEAD * DIM; idx += 256) {
    const int h = idx / DIM, d = idx % DIM;
    float a = 0.0f;
    for (int c = 0; c < NCHUNK; ++c)
      a += __expf(pm[(base + c) * HPAD + h] - Mh[h]) *
           pacc[((base + c) * HPAD + h) * DIM + d];
    ctxRaw[h][d] = a / Lh[h];
  }
  __syncthreads();

  for (int idx = t; idx < NHEAD * DIM; idx += 256) {
    const int h = idx / DIM, e = idx % DIM;
    float s = bv[h * DIM + e];
    for (int d = 0; d < DIM; ++d)
      s += ctxRaw[h][d] * Wv[((size_t)h * DIM + d) * DIM + e];
    x[h * DIM + e] = s;  // head-major feature order == reference dstack
  }
  __syncthreads();

  if (t < HIDW) {
    float s = b1[t];
    for (int i = 0; i < NHEAD * DIM; ++i) s += x[i] * W1[(size_t)i * HIDW + t];
    h1[t] = s > 0.0f ? s : 0.0f;
  }
  __syncthreads();

  if (t == 0) {
    float s = b2[0];
    for (int j = 0; j < HIDW; ++j) s += h1[j] * W2[j];
    out[b] = s;
  }
}

// ---------------------------------------------------------------------------
extern "C" void kernel_launch(void* const* d_in, const int* in_sizes, int n_in,
                              void* d_out, int out_size, void* d_ws,
                              size_t ws_size, hipStream_t stream) {
  const float* enc    = (const float*)d_in[0];
  const float* hidden = (const float*)d_in[1];
  const float* Wq     = (const float*)d_in[2];
  const float* bq     = (const float*)d_in[3];
  const float* Wk     = (const float*)d_in[4];
  const float* bk     = (const float*)d_in[5];
  const float* Wv     = (const float*)d_in[6];
  const float* bv     = (const float*)d_in[7];
  const float* W1     = (const float*)d_in[8];
  const float* b1     = (const float*)d_in[9];
  const float* W2     = (const float*)d_in[10];
  const float* b2     = (const float*)d_in[11];
  float* out = (float*)d_out;

  float* w = (float*)d_ws;
  float* qk    = w;                                        // B*HPAD*DIM
  float* sbias = qk + (size_t)BATCH * HPAD * DIM;          // B*HPAD
  float* pm    = sbias + (size_t)BATCH * HPAD;             // B*NC*HPAD
  float* pl    = pm + (size_t)BATCH * NCHUNK * HPAD;       // B*NC*HPAD
  float* pacc  = pl + (size_t)BATCH * NCHUNK * HPAD;       // B*NC*HPAD*DIM

  prep_kernel<<<BATCH, 64, 0, stream>>>(hidden, Wq, bq, Wk, bk, qk, sbias);
  attn_partial_kernel<<<dim3(NCHUNK, BATCH), 32, 0, stream>>>(enc, qk, sbias,
                                                              pm, pl, pacc);
  finalize_kernel<<<BATCH, 256, 0, stream>>>(pm, pl, pacc, Wv, bv, W1, b1, W2,
                                             b2, out);
}